// SpatialTransformer_14285061226615
// MI455X (gfx1250) — compile-verified
//
#include <hip/hip_runtime.h>

// VoxelMorph 2D SpatialTransformer: out[b,c,h,w] = bilinear(src[b,c], (h,w)+flow[b,:,h,w])
// zeros padding; sample coords = identity grid + flow (align_corners=True semantics).
//
// Shapes fixed by the reference: B=16, C=4, H=W=768.
// Bandwidth-bound: ~378 MB min traffic -> ~16 us at 23.3 TB/s HBM.
//
// Grid does the index decomposition (no integer division in the kernel):
//   blockIdx.x in [0, W/256)  -> w = blockIdx.x*256 + tid  (vector)
//   blockIdx.y in [0, H)      -> h                        (scalar)
//   blockIdx.z in [0, B)      -> b                        (scalar)
// One thread per (b,h,w): bilinear weights computed once, reused across C=4
// channels. Flow reads / out writes are stream-once -> non-temporal; src
// gathers are temporal (151 MB src nearly fits the 192 MB L2, flow is +/-5 px
// so gathers are cacheline-local).

#define B_DIM 16
#define C_DIM 4
#define H_DIM 768
#define W_DIM 768
#define HW (H_DIM * W_DIM)
#define BLOCK 256

__global__ __launch_bounds__(BLOCK) void spatial_transformer_kernel(
    const float* __restrict__ src,   // [B, C, H, W]
    const float* __restrict__ flow,  // [B, 2, H, W]  (plane 0: dy, plane 1: dx)
    float* __restrict__ out)         // [B, C, H, W]
{
    const int w = blockIdx.x * BLOCK + threadIdx.x;  // lane-varying
    const int h = blockIdx.y;                        // scalar
    const int b = blockIdx.z;                        // scalar

    const int hw = h * W_DIM + w;                    // offset within a plane

    // Flow planes are streamed exactly once -> non-temporal loads (TH=NT),
    // keeping L2 dedicated to the heavily re-read src planes.
    const int fbase = b * (2 * HW) + hw;
    const float fy = __builtin_nontemporal_load(flow + fbase);       // dy
    const float fx = __builtin_nontemporal_load(flow + fbase + HW);  // dx

    const float y = (float)h + fy;
    const float x = (float)w + fx;

    const float y0f = floorf(y);
    const float x0f = floorf(x);
    const float wy1 = y - y0f, wy0 = 1.0f - wy1;
    const float wx1 = x - x0f, wx0 = 1.0f - wx1;

    const int iy0 = (int)y0f, iy1 = iy0 + 1;
    const int ix0 = (int)x0f, ix1 = ix0 + 1;

    // Zeros padding: clamp the address, zero the weight when OOB (branchless,
    // identical math to the reference's mask-multiply).
    const float my0 = (iy0 >= 0 && iy0 < H_DIM) ? 1.0f : 0.0f;
    const float my1 = (iy1 >= 0 && iy1 < H_DIM) ? 1.0f : 0.0f;
    const float mx0 = (ix0 >= 0 && ix0 < W_DIM) ? 1.0f : 0.0f;
    const float mx1 = (ix1 >= 0 && ix1 < W_DIM) ? 1.0f : 0.0f;

    const int cy0 = min(max(iy0, 0), H_DIM - 1);
    const int cy1 = min(max(iy1, 0), H_DIM - 1);
    const int cx0 = min(max(ix0, 0), W_DIM - 1);
    const int cx1 = min(max(ix1, 0), W_DIM - 1);

    const float w00 = (wy0 * wx0) * (my0 * mx0);
    const float w01 = (wy0 * wx1) * (my0 * mx1);
    const float w10 = (wy1 * wx0) * (my1 * mx0);
    const float w11 = (wy1 * wx1) * (my1 * mx1);

    // Corner offsets within one [H,W] plane (shared across all channels).
    const int r0  = cy0 * W_DIM;
    const int r1  = cy1 * W_DIM;
    const int o00 = r0 + cx0;
    const int o01 = r0 + cx1;
    const int o10 = r1 + cx0;
    const int o11 = r1 + cx1;

    const int plane0 = b * (C_DIM * HW);  // start of (b, c=0) plane (scalar)

    #pragma unroll
    for (int c = 0; c < C_DIM; ++c) {
        const float* __restrict__ p = src + plane0 + c * HW;
        // Temporal gathers: +/-~5 px displacement -> strong L0/L2 hit rate.
        float v = p[o00] * w00;
        v = fmaf(p[o01], w01, v);
        v = fmaf(p[o10], w10, v);
        v = fmaf(p[o11], w11, v);
        // Output is write-once -> non-temporal store, bypass/rinse L2.
        __builtin_nontemporal_store(v, out + plane0 + c * HW + hw);
    }
}

extern "C" void kernel_launch(void* const* d_in, const int* in_sizes, int n_in,
                              void* d_out, int out_size, void* d_ws, size_t ws_size,
                              hipStream_t stream) {
    (void)in_sizes; (void)n_in; (void)out_size; (void)d_ws; (void)ws_size;
    const float* src  = (const float*)d_in[0];  // [16,4,768,768] fp32
    const float* flow = (const float*)d_in[1];  // [16,2,768,768] fp32
    float* out = (float*)d_out;                 // [16,4,768,768] fp32

    dim3 grid(W_DIM / BLOCK, H_DIM, B_DIM);     // (3, 768, 16) = 36,864 blocks
    dim3 block(BLOCK, 1, 1);                    // 8 wave32 waves per block
    spatial_transformer_kernel<<<grid, block, 0, stream>>>(src, flow, out);
}